// BlazeEarInferenceExportable_40398462386439
// MI455X (gfx1250) — compile-verified
//
#include <hip/hip_runtime.h>
#include <hip/hip_bf16.h>

#define MAX_DET 100
#define CAND_CAP 4096
#define NBINS 256
#define SCALE_F 128.0f
#define CONF_THRESH 0.75f
#define IOU_THRESH 0.3f
#define SCORE_CLIP 100.0f

typedef float v2f __attribute__((ext_vector_type(2)));
typedef float v8f __attribute__((ext_vector_type(8)));

// Monotonic float->uint key: preserves total order of floats as unsigned ints.
__device__ __forceinline__ unsigned key_of(float s) {
    unsigned u = __float_as_uint(s);
    return (u & 0x80000000u) ? ~u : (u | 0x80000000u);
}

// ---------------- K1: 256-bin histogram of key top-8-bits over all scores ----
__global__ void k_hist(const float* __restrict__ scores,
                       unsigned* __restrict__ ghist, int nvec) {
    __shared__ unsigned lh[16 * 257];  // 16 replicas, +1 pad vs bank conflicts
    for (int i = threadIdx.x; i < 16 * 257; i += blockDim.x) lh[i] = 0u;
    __syncthreads();
    const unsigned rep = (threadIdx.x & 15) * 257;
    const float4* s4 = (const float4*)scores;
    const int stride = gridDim.x * blockDim.x;
    for (int i = blockIdx.x * blockDim.x + threadIdx.x; i < nvec; i += stride) {
        float4 v = s4[i];
        atomicAdd(&lh[rep + (key_of(v.x) >> 24)], 1u);
        atomicAdd(&lh[rep + (key_of(v.y) >> 24)], 1u);
        atomicAdd(&lh[rep + (key_of(v.z) >> 24)], 1u);
        atomicAdd(&lh[rep + (key_of(v.w) >> 24)], 1u);
    }
    __syncthreads();
    for (int b = threadIdx.x; b < NBINS; b += blockDim.x) {
        unsigned s = 0u;
        #pragma unroll
        for (int r = 0; r < 16; ++r) s += lh[r * 257 + b];
        if (s) atomicAdd(&ghist[b], s);
    }
}

// ---------------- K2: find highest bin where suffix count >= MAX_DET ---------
__global__ void k_thresh(const unsigned* __restrict__ ghist,
                         unsigned* __restrict__ thr) {
    __shared__ unsigned sp[NBINS];
    const int t = threadIdx.x;
    sp[t] = ghist[NBINS - 1 - t];   // reversed: prefix of sp == suffix of hist
    __syncthreads();
    for (int d = 1; d < NBINS; d <<= 1) {
        unsigned v = (t >= d) ? sp[t - d] : 0u;
        __syncthreads();
        sp[t] += v;
        __syncthreads();
    }
    if (sp[t] >= MAX_DET && (t == 0 || sp[t - 1] < MAX_DET))
        *thr = (unsigned)(NBINS - 1 - t);
    if (t == NBINS - 1 && sp[t] < MAX_DET) *thr = 0u;  // fewer than 100 total
}

// ---------------- K3: compact candidates above threshold bin -----------------
__global__ void k_compact(const float* __restrict__ scores,
                          const unsigned* __restrict__ thrp,
                          unsigned* __restrict__ cnt,
                          unsigned* __restrict__ keys,
                          unsigned* __restrict__ idxs, int nvec) {
    const unsigned thr = *thrp;
    const float4* s4 = (const float4*)scores;
    const int stride = gridDim.x * blockDim.x;
    for (int i = blockIdx.x * blockDim.x + threadIdx.x; i < nvec; i += stride) {
        float4 v = s4[i];
        float e[4] = {v.x, v.y, v.z, v.w};
        #pragma unroll
        for (int c = 0; c < 4; ++c) {
            unsigned k = key_of(e[c]);
            if ((k >> 24) >= thr) {
                unsigned slot = atomicAdd(cnt, 1u);
                if (slot < CAND_CAP) {
                    keys[slot] = k;
                    idxs[slot] = (unsigned)(i * 4 + c);
                }
            }
        }
    }
}

// ---------------- K4: sort candidates, decode top-100, WMMA+NMS, output ------
__global__ void __launch_bounds__(1024)
k_final(const float* __restrict__ raw_boxes, const float* __restrict__ raw_scores,
        const float* __restrict__ anchors, const unsigned* __restrict__ cntp,
        const unsigned* __restrict__ keys, const unsigned* __restrict__ idxs,
        float* __restrict__ out, int n_anchors) {
    __shared__ union SU {
        unsigned long long srt[CAND_CAP]; // phase 1: bitonic sort buffer (32KB)
        float asum[112 * 112];            // phase 2: area[i]+area[j] via WMMA (49KB)
    } u;
    __shared__ float sbox[MAX_DET][4];
    __shared__ float sscore[MAX_DET];
    __shared__ float sarea[112];
    __shared__ int skeep[MAX_DET];
    __shared__ int sidx[MAX_DET];

    const int tid = threadIdx.x;
    unsigned cnt = *cntp;
    if (cnt > CAND_CAP) cnt = CAND_CAP;

    // Composite key: (score key desc, index asc) -> matches top_k tie-breaking.
    for (int i = tid; i < CAND_CAP; i += blockDim.x) {
        unsigned long long e = 0ull;
        if (i < (int)cnt)
            e = ((unsigned long long)keys[i] << 32) | (unsigned)(~idxs[i]);
        u.srt[i] = e;
    }
    __syncthreads();

    // Bitonic sort, descending, 4096 elements with 1024 threads.
    for (int k = 2; k <= CAND_CAP; k <<= 1)
        for (int j = k >> 1; j > 0; j >>= 1) {
            for (int t = tid; t < CAND_CAP; t += blockDim.x) {
                int x = t ^ j;
                if (x > t) {
                    unsigned long long a = u.srt[t], b = u.srt[x];
                    bool up = ((t & k) == 0);
                    if (up ? (a < b) : (a > b)) { u.srt[t] = b; u.srt[x] = a; }
                }
            }
            __syncthreads();
        }

    // Extract top-100 indices.
    const int m = (cnt < (unsigned)MAX_DET) ? (int)cnt : MAX_DET;
    if (tid < MAX_DET) {
        if (tid < m) {
            unsigned long long e = u.srt[tid];
            sidx[tid] = (int)(~(unsigned)(e & 0xffffffffull));
        } else {
            sidx[tid] = -1;
        }
        skeep[tid] = 1;
    }
    __syncthreads();  // after this, u.srt is dead; u.asum may be written

    // Gather + decode only the 100 winning anchors.
    if (tid < 112) sarea[tid] = 0.0f;
    if (tid < MAX_DET) {
        int idx = sidx[tid];
        if (idx >= 0 && idx < n_anchors) {
            float rs = raw_scores[idx];
            rs = fminf(fmaxf(rs, -SCORE_CLIP), SCORE_CLIP);
            sscore[tid] = 1.0f / (1.0f + expf(-rs));
            const float4 rb = ((const float4*)raw_boxes)[idx];
            const float4 an = ((const float4*)anchors)[idx];
            float xc = rb.x * (1.0f / SCALE_F) * an.z + an.x;
            float yc = rb.y * (1.0f / SCALE_F) * an.w + an.y;
            float w  = rb.z * (1.0f / SCALE_F) * an.z;
            float h  = rb.w * (1.0f / SCALE_F) * an.w;
            float ymin = yc - 0.5f * h, ymax = yc + 0.5f * h;
            float xmin = xc - 0.5f * w, xmax = xc + 0.5f * w;
            float b0 = fminf(ymin, ymax), b1 = fminf(xmin, xmax);
            float b2 = fmaxf(ymin, ymax), b3 = fmaxf(xmin, xmax);
            sbox[tid][0] = b0; sbox[tid][1] = b1;
            sbox[tid][2] = b2; sbox[tid][3] = b3;
            sarea[tid] = (b2 - b0) * (b3 - b1);
        } else {
            sscore[tid] = 0.0f;
            sbox[tid][0] = sbox[tid][1] = sbox[tid][2] = sbox[tid][3] = 0.0f;
            sarea[tid] = 0.0f;
        }
    }
    __syncthreads();

    // Pairwise union base area[i]+area[j] as a rank-2 outer product on the
    // matrix core: A(16x4) = [area | 1 | 0 | 0], B(4x16) = [1 ; area ; 0 ; 0].
    // 7x7 tiles of 16x16 cover the padded 112x112 matrix; tiles strided over the 32 waves (waves 0-16 do two).
    {
        const int wave = tid >> 5;
        const int lane = tid & 31;
        for (int t = wave; t < 49; t += 32) {
            const int r0 = (t / 7) * 16, c0 = (t % 7) * 16;
            const bool lo = lane < 16;          // lanes 0-15 hold K=0,1; hi lanes K=2,3 (zero)
            v2f a, b;
            a.x = lo ? sarea[r0 + (lane & 15)] : 0.0f;  // A[:,0] = area_i
            a.y = lo ? 1.0f : 0.0f;                      // A[:,1] = 1
            b.x = lo ? 1.0f : 0.0f;                      // B[0,:] = 1
            b.y = lo ? sarea[c0 + (lane & 15)] : 0.0f;   // B[1,:] = area_j
            v8f c = {0.0f, 0.0f, 0.0f, 0.0f, 0.0f, 0.0f, 0.0f, 0.0f};
            v8f d = __builtin_amdgcn_wmma_f32_16x16x4_f32(
                false, a, false, b, (short)0, c, false, false);
            const int col = c0 + (lane & 15);
            const int rbase = r0 + (lo ? 0 : 8);   // C/D layout: hi lanes hold M=8..15
            #pragma unroll
            for (int r = 0; r < 8; ++r)
                u.asum[(rbase + r) * 112 + col] = d[r];
        }
    }
    __syncthreads();

    // Greedy NMS: sequential over i, parallel over j.
    for (int i = 0; i < MAX_DET - 1; ++i) {
        __syncthreads();
        if (skeep[i]) {
            int j = tid;
            if (j > i && j < MAX_DET && skeep[j]) {
                float iy = fminf(sbox[i][2], sbox[j][2]) - fmaxf(sbox[i][0], sbox[j][0]);
                float ix = fminf(sbox[i][3], sbox[j][3]) - fmaxf(sbox[i][1], sbox[j][1]);
                iy = fmaxf(iy, 0.0f); ix = fmaxf(ix, 0.0f);
                float inter = iy * ix;
                float uni = fmaxf(u.asum[i * 112 + j] - inter, 1e-9f);
                if (inter / uni > IOU_THRESH) skeep[j] = 0;
            }
        }
    }
    __syncthreads();

    // Stable valid-first compaction; zero the rest.
    for (int i = tid; i < MAX_DET * 5; i += blockDim.x) out[i] = 0.0f;
    __syncthreads();
    if (tid == 0) {
        int pos = 0;
        for (int j = 0; j < MAX_DET; ++j) {
            if (skeep[j] && sscore[j] >= CONF_THRESH) {
                out[pos * 5 + 0] = sbox[j][0];
                out[pos * 5 + 1] = sbox[j][1];
                out[pos * 5 + 2] = sbox[j][2];
                out[pos * 5 + 3] = sbox[j][3];
                out[pos * 5 + 4] = sscore[j];
                ++pos;
            }
        }
    }
}

extern "C" void kernel_launch(void* const* d_in, const int* in_sizes, int n_in,
                              void* d_out, int out_size, void* d_ws, size_t ws_size,
                              hipStream_t stream) {
    const float* raw_boxes  = (const float*)d_in[0];
    const float* raw_scores = (const float*)d_in[1];
    const float* anchors    = (const float*)d_in[2];
    float* out = (float*)d_out;
    const int n = in_sizes[1];  // raw_scores flat count == N_ANCHORS

    // Workspace layout (u32 units):
    // [0..255] hist | [256] candCount | [257] thrBin | [260..] keys | [260+CAP..] idxs
    unsigned* ws   = (unsigned*)d_ws;
    unsigned* hist = ws;
    unsigned* cnt  = ws + 256;
    unsigned* thr  = ws + 257;
    unsigned* keys = ws + 260;
    unsigned* idxs = ws + 260 + CAND_CAP;

    hipMemsetAsync(d_ws, 0, (256 + 4) * sizeof(unsigned), stream);

    const int nvec = n / 4;  // N is a multiple of 4
    k_hist   <<<512, 256, 0, stream>>>(raw_scores, hist, nvec);
    k_thresh <<<1, NBINS, 0, stream>>>(hist, thr);
    k_compact<<<512, 256, 0, stream>>>(raw_scores, thr, cnt, keys, idxs, nvec);
    k_final  <<<1, 1024, 0, stream>>>(raw_boxes, raw_scores, anchors,
                                      cnt, keys, idxs, out, n);
}